// VectorQuantizer_47717086658939
// MI455X (gfx1250) — compile-verified
//
#include <hip/hip_runtime.h>

typedef __attribute__((ext_vector_type(2))) float v2f;
typedef __attribute__((ext_vector_type(8))) float v8f;

#define DDIM 64   // embedding dim
#define KCODES 16 // codebook size

__global__ void vq_init_loss(float* loss) {
    if (threadIdx.x == 0 && blockIdx.x == 0) *loss = 0.0f;
}

static __device__ __forceinline__ unsigned ballot32(bool p) {
#if __has_builtin(__builtin_amdgcn_ballot_w32)
    return __builtin_amdgcn_ballot_w32(p);
#else
    return (unsigned)__ballot(p);
#endif
}

// monotonic float -> u32 key (order-preserving for all finite values)
static __device__ __forceinline__ unsigned fkey(float f) {
    unsigned u = __float_as_uint(f);
    return u ^ (unsigned)(((int)u >> 31) | 0x80000000);
}
static __device__ __forceinline__ float funkey(unsigned k) {
    unsigned u = (k & 0x80000000u) ? (k ^ 0x80000000u) : ~k;
    return __uint_as_float(u);
}

__global__ __launch_bounds__(256) void vq_wmma_kernel(
    const float* __restrict__ z,
    const float* __restrict__ emb,
    float* __restrict__ qout,
    float* __restrict__ idxout,
    float* __restrict__ lossout,
    int nrows)
{
    const int lane = threadIdx.x & 31;
    const int g    = lane >> 4;   // half-wave: 0 or 1
    const int gl   = lane & 15;   // lane within half
    const int waves_per_block = blockDim.x >> 5;
    const int wave_global = blockIdx.x * waves_per_block + (threadIdx.x >> 5);
    const int nwaves = gridDim.x * waves_per_block;
    const int ntiles = nrows >> 4;

    // ---- hoisted B-fragments: embedding codes as matrix columns ----
    // B (4x16, f32): VGPR j, half g holds K = kbase + 2*g + j, N = gl
    v2f bfrag[16];
    #pragma unroll
    for (int kk = 0; kk < 16; ++kk) {
        bfrag[kk] = *(const v2f*)(emb + gl * DDIM + kk * 4 + 2 * g);
    }
    // ||e_n||^2 for n = gl (same in both halves)
    float enorm = 0.0f;
    #pragma unroll
    for (int d = 0; d < DDIM; d += 4) {
        float4 e = *(const float4*)(emb + gl * DDIM + d);
        enorm += e.x * e.x + e.y * e.y + e.z * e.z + e.w * e.w;
    }

    // incremented base pointers: no per-tile 64-bit address math
    const size_t tile_stride = (size_t)nwaves * 16 * DDIM;        // floats
    const float* zp = z    + (size_t)wave_global * 16 * DDIM + (size_t)gl * DDIM + 2 * g;
    float*       qp = qout + (size_t)wave_global * 16 * DDIM + gl * 4;
    float*       ip = idxout + (size_t)wave_global * 16;

    float lsum = 0.0f;

    for (int t = wave_global; t < ntiles;
         t += nwaves, zp += tile_stride, qp += tile_stride, ip += (size_t)nwaves * 16) {

        // ---- stage all 16 A-fragments: 16 independent b64 loads (one clause) ----
        v2f a[16];
        #pragma unroll
        for (int kk = 0; kk < 16; ++kk) {
            a[kk] = *(const v2f*)(zp + kk * 4);
        }

        // ---- Gram block: 16 chained f32 WMMAs; ||z_row||^2 from registers ----
        v8f acc = {};
        float zn0 = 0.0f, zn1 = 0.0f, zn2 = 0.0f, zn3 = 0.0f;
        #pragma unroll
        for (int kk = 0; kk < 16; ++kk) {
            acc = __builtin_amdgcn_wmma_f32_16x16x4_f32(
                false, a[kk], false, bfrag[kk], (short)0, acc, false, false);
        }
        #pragma unroll
        for (int kk = 0; kk < 16; kk += 4) {
            zn0 = fmaf(a[kk + 0].x, a[kk + 0].x, fmaf(a[kk + 0].y, a[kk + 0].y, zn0));
            zn1 = fmaf(a[kk + 1].x, a[kk + 1].x, fmaf(a[kk + 1].y, a[kk + 1].y, zn1));
            zn2 = fmaf(a[kk + 2].x, a[kk + 2].x, fmaf(a[kk + 2].y, a[kk + 2].y, zn2));
            zn3 = fmaf(a[kk + 3].x, a[kk + 3].x, fmaf(a[kk + 3].y, a[kk + 3].y, zn3));
        }
        float zn = (zn0 + zn1) + (zn2 + zn3);
        // combine k-coverage of the two halves -> full ||z_{m0+gl}||^2 in every lane
        zn += __shfl_xor(zn, 16, 32);

        // ---- per-row argmin over 16 codes: branchless u32 key min-reduce ----
        #pragma unroll
        for (int r = 0; r < 8; ++r) {
            // score = ||e_n||^2 - 2 z.e_n  (row-constant ||z||^2 dropped for argmin)
            float s = fmaf(-2.0f, acc[r], enorm);
            unsigned key  = fkey(s);
            unsigned kmin = key;
            #pragma unroll
            for (int off = 8; off >= 1; off >>= 1) {
                unsigned ok = __shfl_xor(kmin, off, 32);
                kmin = (ok < kmin) ? ok : kmin;   // v_min_u32, stays within half
            }
            // winner = lowest lane in this half whose key equals the min
            unsigned bal   = ballot32(key == kmin);
            unsigned gmask = (bal >> (16 * g)) & 0xFFFFu;
            const int idx  = __ffs(gmask) - 1;    // argmin tie-break (lowest n)

            // quantized row write: 16 lanes x float4 = one contiguous 256B row
            float4 e = *(const float4*)(emb + idx * DDIM + gl * 4);
            *(float4*)(qp + (r + 8 * g) * DDIM) = e;
            if (gl == 0) ip[r + 8 * g] = (float)idx;

            // ||z - e*||^2 = ||z||^2 + min_score ; one lane per half owns row m
            lsum += (gl == r + 8 * g) ? (zn + funkey(kmin)) : 0.0f;
        }
    }

    // ---- wave-reduce loss, one atomic per wave ----
    #pragma unroll
    for (int off = 16; off >= 1; off >>= 1)
        lsum += __shfl_xor(lsum, off, 32);
    if (lane == 0) {
        const float scale = 1.25f / ((float)nrows * (float)DDIM);
        atomicAdd(lossout, lsum * scale);
    }
}

extern "C" void kernel_launch(void* const* d_in, const int* in_sizes, int n_in,
                              void* d_out, int out_size, void* d_ws, size_t ws_size,
                              hipStream_t stream) {
    const float* z   = (const float*)d_in[0];   // [B, 64]
    const float* emb = (const float*)d_in[1];   // [16, 64]
    const int nrows = in_sizes[0] / DDIM;       // B = 1048576

    float* qout    = (float*)d_out;                   // [B, 64]
    float* idxout  = qout + (size_t)nrows * DDIM;     // [B]
    float* lossout = idxout + nrows;                  // [1]

    vq_init_loss<<<1, 1, 0, stream>>>(lossout);
    vq_wmma_kernel<<<2048, 256, 0, stream>>>(z, emb, qout, idxout, lossout, nrows);
}